// MyContrastiveLoss_77558519432094
// MI455X (gfx1250) — compile-verified
//
#include <hip/hip_runtime.h>
#include <hip/hip_bf16.h>
#include <stdint.h>

// ---------------------------------------------------------------------------
// MyContrastiveLoss on MI455X (gfx1250, wave32).
//   out = CE(post, labels) + 0.01 * sum_i sqrt(||pre[i] - pre[pair[i]]||^2)
// pair[] reproduces jax's _pair_indexes: argmax over threefry2x32-based
// uniform noise (key(42)), masked by same-label & !=self, first-occurrence
// tie-break, fallback to self.
// ---------------------------------------------------------------------------

typedef __attribute__((ext_vector_type(2))) float v2f;
typedef __attribute__((ext_vector_type(8))) float v8f;

// ---------------- Threefry2x32 (JAX classic counter scheme, seed 42) -------
__device__ __forceinline__ uint32_t rotl32(uint32_t v, uint32_t n) {
  return (v << n) | (v >> (32u - n));
}

__device__ __forceinline__ void threefry2x32_42(uint32_t x0, uint32_t x1,
                                                uint32_t& o0, uint32_t& o1) {
  const uint32_t ks0 = 0u;                      // hi(seed=42)
  const uint32_t ks1 = 42u;                     // lo(seed=42)
  const uint32_t ks2 = ks0 ^ ks1 ^ 0x1BD11BDAu;
  x0 += ks0; x1 += ks1;
#define TF_R(r) { x0 += x1; x1 = rotl32(x1, r); x1 ^= x0; }
  TF_R(13) TF_R(15) TF_R(26) TF_R(6)
  x0 += ks1; x1 += ks2 + 1u;
  TF_R(17) TF_R(29) TF_R(16) TF_R(24)
  x0 += ks2; x1 += ks0 + 2u;
  TF_R(13) TF_R(15) TF_R(26) TF_R(6)
  x0 += ks0; x1 += ks1 + 3u;
  TF_R(17) TF_R(29) TF_R(16) TF_R(24)
  x0 += ks1; x1 += ks2 + 4u;
  TF_R(13) TF_R(15) TF_R(26) TF_R(6)
  x0 += ks2; x1 += ks0 + 5u;
#undef TF_R
  o0 = x0; o1 = x1;
}

__device__ __forceinline__ float bits_to_uniform(uint32_t bits) {
  // jax.random.uniform: bitcast((bits>>9)|0x3f800000) - 1, clamped at 0
  return fmaxf(__uint_as_float((bits >> 9) | 0x3f800000u) - 1.0f, 0.0f);
}

// ---------------- Kernel 0: zero the scalar accumulators -------------------
__global__ void k_init(float* accums) {
  accums[0] = 0.0f;  // distance sum
  accums[1] = 0.0f;  // CE sum
}

// ---------------- Kernel 1: pair indexes -----------------------------------
// One block handles rows r0=blockIdx.x and r1=r0+B/2: a single threefry call
// with counters (r0*B+c, r1*B+c) yields the noise for BOTH rows at column c
// (classic jax scheme splits the flat iota in halves). 256 threads stride
// columns; LDS tree-reduction implements first-occurrence argmax.
__global__ void __launch_bounds__(256)
k_pair(const int* __restrict__ labels, int* __restrict__ pair, int B) {
  __shared__ int   slab[8192];
  __shared__ float sv[256];
  __shared__ int   si[256];

  const int tid = threadIdx.x;
  const int r0  = blockIdx.x;
  const int r1  = r0 + (B >> 1);

  for (int i = tid; i < B; i += 256) slab[i] = labels[i];
  __syncthreads();

  const int lab0 = slab[r0];
  const int lab1 = slab[r1];
  const uint32_t base0 = (uint32_t)r0 * (uint32_t)B;
  const uint32_t base1 = (uint32_t)r1 * (uint32_t)B;

  float best0 = -1.0f, best1 = -1.0f;
  int   bi0 = 0x7fffffff, bi1 = 0x7fffffff;

  for (int c = tid; c < B; c += 256) {
    uint32_t o0, o1;
    threefry2x32_42(base0 + (uint32_t)c, base1 + (uint32_t)c, o0, o1);
    const int lc = slab[c];
    float s0 = (lc == lab0 && c != r0) ? bits_to_uniform(o0) : -1.0f;
    float s1 = (lc == lab1 && c != r1) ? bits_to_uniform(o1) : -1.0f;
    if (s0 > best0) { best0 = s0; bi0 = c; }  // strict > keeps earliest c
    if (s1 > best1) { best1 = s1; bi1 = c; }
  }

  // Reduce row r0: higher value wins; ties -> lower index (argmax semantics).
  sv[tid] = best0; si[tid] = bi0; __syncthreads();
  for (int s = 128; s > 0; s >>= 1) {
    if (tid < s) {
      float v = sv[tid + s]; int ix = si[tid + s];
      if (v > sv[tid] || (v == sv[tid] && ix < si[tid])) { sv[tid] = v; si[tid] = ix; }
    }
    __syncthreads();
  }
  if (tid == 0) pair[r0] = (sv[0] >= 0.0f) ? si[0] : r0;
  __syncthreads();

  // Reduce row r1.
  sv[tid] = best1; si[tid] = bi1; __syncthreads();
  for (int s = 128; s > 0; s >>= 1) {
    if (tid < s) {
      float v = sv[tid + s]; int ix = si[tid + s];
      if (v > sv[tid] || (v == sv[tid] && ix < si[tid])) { sv[tid] = v; si[tid] = ix; }
    }
    __syncthreads();
  }
  if (tid == 0) pair[r1] = (sv[0] >= 0.0f) ? si[0] : r1;
}

// ---------------- Kernel 2: cross-entropy ----------------------------------
// One wave32 per row; lanes stride the 50 classes; shuffle reductions.
__global__ void __launch_bounds__(256)
k_ce(const float* __restrict__ post, const int* __restrict__ labels,
     float* __restrict__ ce_accum, int B, int C) {
  const int lane = threadIdx.x & 31;
  const int wave = threadIdx.x >> 5;
  const int row  = blockIdx.x * (blockDim.x >> 5) + wave;
  if (row >= B) return;

  const float* x = post + (size_t)row * (size_t)C;
  float m = -3.402823466e38f;
  for (int c = lane; c < C; c += 32) m = fmaxf(m, x[c]);
  for (int o = 16; o > 0; o >>= 1) m = fmaxf(m, __shfl_xor(m, o, 32));
  float s = 0.0f;
  for (int c = lane; c < C; c += 32) s += expf(x[c] - m);
  for (int o = 16; o > 0; o >>= 1) s += __shfl_xor(s, o, 32);
  if (lane == 0) {
    const float lse = m + logf(s);
    atomicAdd(ce_accum, lse - x[labels[row]]);
  }
}

// ---------------- Kernel 3: paired distance via WMMA -----------------------
// One wave32 per 16-row tile. A-fragment = diff^2 for a 16x4 chunk (ISA
// 32-bit A layout: lanes 0-15 hold K={0,1}, lanes 16-31 hold K={2,3}),
// B = all-ones 4x16, so D = A*1 + C accumulates exact f32 row sums on the
// matrix pipe. After the K loop, C/D layout puts rows M=0..7 in lane 0's
// 8 accumulator VGPRs and M=8..15 in lane 16's.
__global__ void __launch_bounds__(256)
k_dist(const float* __restrict__ pre, const int* __restrict__ pair,
       float* __restrict__ dist_accum, int D) {
  const int lane = threadIdx.x & 31;
  const int wave = threadIdx.x >> 5;
  const int tile = blockIdx.x * (blockDim.x >> 5) + wave;  // 16-row tile

  const int m    = lane & 15;
  const int row  = tile * 16 + m;
  const int prow = pair[row];
  const float* pA = pre + (size_t)row  * (size_t)D;
  const float* pB = pre + (size_t)prow * (size_t)D;
  const int koff = (lane >> 4) * 2;  // lanes 0-15 -> K 0,1 ; lanes 16-31 -> K 2,3

  const v2f ones = {1.0f, 1.0f};     // all-ones B works for any B layout
  v8f acc = {};

  for (int k = 0; k < D; k += 4) {
    float2 a = *(const float2*)(pA + k + koff);
    float2 b = *(const float2*)(pB + k + koff);
    float d0 = a.x - b.x;
    float d1 = a.y - b.y;
    v2f afrag = {d0 * d0, d1 * d1};
    // 8 args: (neg_a, A, neg_b, B, c_mod, C, reuse_a, reuse_b)
    acc = __builtin_amdgcn_wmma_f32_16x16x4_f32(
        false, afrag, false, ones, (short)0, acc, false, false);
  }

  if (lane == 0 || lane == 16) {     // hold rows M=0..7 / M=8..15 respectively
    float s = 0.0f;
#pragma unroll
    for (int j = 0; j < 8; ++j) {
      float ss = acc[j];
      s += sqrtf(ss == 0.0f ? 1e-5f : ss);
    }
    atomicAdd(dist_accum, s);
  }
}

// ---------------- Kernel 4: combine ----------------------------------------
__global__ void k_final(const float* __restrict__ accums, float* __restrict__ out,
                        int B) {
  out[0] = accums[1] / (float)B + 0.01f * accums[0];
}

// ---------------------------------------------------------------------------
extern "C" void kernel_launch(void* const* d_in, const int* in_sizes, int n_in,
                              void* d_out, int out_size, void* d_ws, size_t ws_size,
                              hipStream_t stream) {
  (void)n_in; (void)out_size; (void)ws_size;
  const float* pre    = (const float*)d_in[0];  // [B, D]
  const float* post   = (const float*)d_in[1];  // [B, C]
  const int*   labels = (const int*)d_in[2];    // [B]
  float* out = (float*)d_out;

  const int B = in_sizes[2];            // 8192
  const int D = in_sizes[0] / B;        // 2048
  const int C = in_sizes[1] / B;        // 50

  int*   pair   = (int*)d_ws;                                   // B ints
  float* accums = (float*)((char*)d_ws + (size_t)B * sizeof(int));  // [dist, ce]

  k_init<<<1, 1, 0, stream>>>(accums);
  k_pair<<<B / 2, 256, 0, stream>>>(labels, pair, B);
  k_ce<<<B / 8, 256, 0, stream>>>(post, labels, accums + 1, B, C);
  k_dist<<<B / 16 / 8, 256, 0, stream>>>(pre, pair, accums, D);
  k_final<<<1, 1, 0, stream>>>(accums, out, B);
}